// PNA_78125455114597
// MI455X (gfx1250) — compile-verified
//
#include <hip/hip_runtime.h>
#include <hip/hip_bf16.h>
#include <math.h>

#define N_NODES 50000
#define N_EDGES 800000
#define DFEAT   64
#define EPS     1e-5f

typedef float v2f __attribute__((ext_vector_type(2)));
typedef float v8f __attribute__((ext_vector_type(8)));

#if defined(__has_builtin)
#if __has_builtin(__builtin_amdgcn_wmma_f32_16x16x4_f32)
#define HAVE_WMMA_F32X4 1
#endif
#endif

// Order-preserving float <-> uint encoding so atomicMax/Min on u32 implement
// exact f32 segment max/min.
__device__ __forceinline__ unsigned encf(float f) {
    unsigned u = __float_as_uint(f);
    return (u & 0x80000000u) ? ~u : (u | 0x80000000u);
}
__device__ __forceinline__ float decf(unsigned u) {
    return (u & 0x80000000u) ? __uint_as_float(u & 0x7FFFFFFFu)
                             : __uint_as_float(~u);
}

// ---------------------------------------------------------------- init ------
__global__ void pna_init(float* __restrict__ sum, float* __restrict__ sumsq,
                         unsigned* __restrict__ umax, unsigned* __restrict__ umin,
                         float* __restrict__ deg, int nd, int n) {
    int i = blockIdx.x * blockDim.x + threadIdx.x;
    if (i < nd) {
        sum[i]   = 0.0f;
        sumsq[i] = 0.0f;
        umax[i]  = 0u;           // < encf(any finite value)
        umin[i]  = 0xFFFFFFFFu;  // > encf(any finite value)
    }
    if (i < n) deg[i] = 0.0f;
}

// ------------------------------------------------------- edge scatter -------
// One thread per (edge, feature). x (12.8MB) and the 51MB of accumulators are
// L2-resident (192MB L2), so gathers + atomics resolve in L2.
__global__ void pna_edge(const float* __restrict__ x, const int* __restrict__ ei,
                         float* __restrict__ deg, float* __restrict__ sum,
                         float* __restrict__ sumsq, unsigned* __restrict__ umax,
                         unsigned* __restrict__ umin) {
    int gid = blockIdx.x * blockDim.x + threadIdx.x;
    int e = gid >> 6;
    if (e >= N_EDGES) return;
    int d = gid & 63;
    int s = ei[e];            // edge_index[0][e]
    int t = ei[N_EDGES + e];  // edge_index[1][e]
    float v = x[s * DFEAT + d];
    int o = t * DFEAT + d;
    atomicAdd(&sum[o], v);
    atomicAdd(&sumsq[o], v * v);
    unsigned enc = encf(v);
    atomicMax(&umax[o], enc);
    atomicMin(&umin[o], enc);
    if (d == 0) atomicAdd(&deg[t], 1.0f);
}

// ------------------------------------------- deterministic logd reduction ---
__global__ void pna_logd_partial(const float* __restrict__ deg,
                                 float* __restrict__ partials, int n) {
    __shared__ float sm[256];
    int tid = threadIdx.x;
    int i = blockIdx.x * 256 + tid;
    sm[tid] = (i < n) ? logf(deg[i] + 1.0f) : 0.0f;
    __syncthreads();
    for (int st = 128; st > 0; st >>= 1) {
        if (tid < st) sm[tid] += sm[tid + st];
        __syncthreads();
    }
    if (tid == 0) partials[blockIdx.x] = sm[0];
}

__global__ void pna_logd_final(const float* __restrict__ partials, int npart,
                               float* __restrict__ deltap, int n) {
    __shared__ float sm[256];
    int tid = threadIdx.x;
    float a = 0.0f;
    for (int i = tid; i < npart; i += 256) a += partials[i];
    sm[tid] = a;
    __syncthreads();
    for (int st = 128; st > 0; st >>= 1) {
        if (tid < st) sm[tid] += sm[tid + st];
        __syncthreads();
    }
    if (tid == 0) *deltap = sm[0] / (float)n;
}

// ---------------------------------- fused agg-assembly + WMMA f32 GEMM ------
// One wave (32 lanes) per 16-node tile. Stage A builds agg[16][256] in LDS
// (mean | max | min | std) plus amp/att scalers. Stage B runs the 768x64 GEMM
// with V_WMMA_F32_16X16X4_F32: 3 scaler blocks x 64 K-steps, 4 column tiles
// held in 4 accumulators so the A fragment is built once per K step.
__global__ void __launch_bounds__(32)
pna_gemm(const float* __restrict__ deg, const float* __restrict__ sum,
         const float* __restrict__ sumsq, const unsigned* __restrict__ umax,
         const unsigned* __restrict__ umin, const float* __restrict__ deltap,
         const float* __restrict__ W, const float* __restrict__ bias,
         float* __restrict__ out) {
    __shared__ float aggL[16 * 256];
    __shared__ float ampL[16];
    __shared__ float attL[16];

    const int tile = blockIdx.x;       // 3125 tiles, exact (50000/16)
    const int lane = threadIdx.x;
    const float delta = *deltap;

    // ---- Stage A: per-node aggregation tile into LDS ----
    for (int m = 0; m < 16; ++m) {
        int node = tile * 16 + m;
        float dg = deg[node];
        float degc = fmaxf(dg, 1.0f);
        bool has = dg > 0.0f;
        if (lane == 0) {
            float ld = logf(dg + 1.0f);
            ampL[m] = ld / delta;
            attL[m] = delta / (ld > 0.0f ? ld : 1.0f);
        }
        #pragma unroll
        for (int h = 0; h < 2; ++h) {
            int d = h * 32 + lane;
            int o = node * DFEAT + d;
            float mean = sum[o] / degc;
            float sq = sumsq[o] / degc;
            float var = sq - mean * mean;
            var = var > 0.0f ? var : 0.0f;       // relu
            float sd = sqrtf(var + EPS);
            float mx = has ? decf(umax[o]) : 0.0f;
            float mn = has ? decf(umin[o]) : 0.0f;
            aggL[m * 256 + d]        = mean;
            aggL[m * 256 + 64 + d]   = mx;
            aggL[m * 256 + 128 + d]  = mn;
            aggL[m * 256 + 192 + d]  = sd;
        }
    }
    __syncthreads();

    const int m16 = lane & 15;
    const int koff = (lane >> 4) * 2;   // lanes 0-15: K=0,1 ; lanes 16-31: K=2,3
    const float myAmp = ampL[m16];
    const float myAtt = attL[m16];

#ifdef HAVE_WMMA_F32X4
    v8f acc0 = {0,0,0,0,0,0,0,0};
    v8f acc1 = {0,0,0,0,0,0,0,0};
    v8f acc2 = {0,0,0,0,0,0,0,0};
    v8f acc3 = {0,0,0,0,0,0,0,0};

    for (int s = 0; s < 3; ++s) {
        const float scal = (s == 0) ? 1.0f : ((s == 1) ? myAmp : myAtt);
        const float* aggRow = &aggL[m16 * 256];
        for (int kb = 0; kb < 64; ++kb) {
            const int p = kb * 4 + koff;          // index within 256-block
            v2f av;
            av.x = aggRow[p] * scal;
            av.y = aggRow[p + 1] * scal;
            const int krow = s * 256 + p;         // global K row in W
            const float* w0 = W + (size_t)krow * DFEAT + m16;
            const float* w1 = w0 + DFEAT;
            v2f bv;
            bv.x = w0[0];  bv.y = w1[0];
            acc0 = __builtin_amdgcn_wmma_f32_16x16x4_f32(false, av, false, bv,
                                                         (short)0, acc0, false, false);
            bv.x = w0[16]; bv.y = w1[16];
            acc1 = __builtin_amdgcn_wmma_f32_16x16x4_f32(false, av, false, bv,
                                                         (short)0, acc1, false, false);
            bv.x = w0[32]; bv.y = w1[32];
            acc2 = __builtin_amdgcn_wmma_f32_16x16x4_f32(false, av, false, bv,
                                                         (short)0, acc2, false, false);
            bv.x = w0[48]; bv.y = w1[48];
            acc3 = __builtin_amdgcn_wmma_f32_16x16x4_f32(false, av, false, bv,
                                                         (short)0, acc3, false, false);
        }
    }

    // Epilogue: C/D layout — VGPR r: lanes 0-15 -> M=r, lanes 16-31 -> M=r+8;
    // N = lane & 15 within each 16-column tile.
    const int rbase = (lane < 16) ? 0 : 8;
    #pragma unroll
    for (int nt = 0; nt < 4; ++nt) {
        const int col = nt * 16 + m16;
        const float bcol = bias[col];
        const v8f acc = (nt == 0) ? acc0 : (nt == 1) ? acc1 : (nt == 2) ? acc2 : acc3;
        #pragma unroll
        for (int r = 0; r < 8; ++r) {
            int node = tile * 16 + rbase + r;
            out[(size_t)node * DFEAT + col] = acc[r] + bcol;
        }
    }
#else
    // Scalar fallback (used only if the fp32 WMMA builtin is unavailable).
    const int half = lane >> 4;
    for (int cc = 0; cc < 32; ++cc) {
        const int col = half * 32 + cc;
        float accv = bias[col];
        for (int s = 0; s < 3; ++s) {
            const float scal = (s == 0) ? 1.0f : ((s == 1) ? myAmp : myAtt);
            for (int q = 0; q < 256; ++q)
                accv += aggL[m16 * 256 + q] * scal * W[(size_t)(s * 256 + q) * DFEAT + col];
        }
        out[(size_t)(tile * 16 + m16) * DFEAT + col] = accv;
    }
#endif
}

// ---------------------------------------------------------------- launch ----
extern "C" void kernel_launch(void* const* d_in, const int* in_sizes, int n_in,
                              void* d_out, int out_size, void* d_ws, size_t ws_size,
                              hipStream_t stream) {
    const float* x  = (const float*)d_in[0];
    const int*   ei = (const int*)d_in[1];
    const float* W  = (const float*)d_in[2];
    const float* b  = (const float*)d_in[3];
    float* out = (float*)d_out;

    const int N = N_NODES, E = N_EDGES;
    const size_t ND = (size_t)N * DFEAT;

    float*    sum      = (float*)d_ws;
    float*    sumsq    = sum + ND;
    unsigned* umax     = (unsigned*)(sumsq + ND);
    unsigned* umin     = umax + ND;
    float*    deg      = (float*)(umin + ND);
    float*    partials = deg + N;
    float*    deltap   = partials + 256;

    const int npartial = (N + 255) / 256;  // 196

    pna_init<<<(int)((ND + 255) / 256), 256, 0, stream>>>(sum, sumsq, umax, umin,
                                                          deg, (int)ND, N);
    pna_edge<<<(E * DFEAT) / 256, 256, 0, stream>>>(x, ei, deg, sum, sumsq,
                                                    umax, umin);
    pna_logd_partial<<<npartial, 256, 0, stream>>>(deg, partials, N);
    pna_logd_final<<<1, 256, 0, stream>>>(partials, npartial, deltap, N);
    pna_gemm<<<N / 16, 32, 0, stream>>>(deg, sum, sumsq, umax, umin, deltap,
                                        W, b, out);
}